// Node3DEmbeddingv2_6949257085356
// MI455X (gfx1250) — compile-verified
//
#include <hip/hip_runtime.h>
#include <hip/hip_bf16.h>
#include <math.h>

typedef __attribute__((ext_vector_type(2))) float v2f;
typedef __attribute__((ext_vector_type(8))) float v8f;

#define B_     2
#define N_     768
#define K_     128
#define INTER_ 256
#define E_     512
#define ROWS_  (B_ * N_)   // 1536

__device__ __forceinline__ float gelu_erf(float x) {
    // exact-erf GELU as in F.gelu default
    return 0.5f * x * (1.0f + erff(x * 0.70710678118654752f));
}

// ---------------------------------------------------------------------------
// Kernel 1: Gaussian basis sum over neighbor axis.
// One block per (b,i) row, 128 threads (one per gaussian k).
// dist_ij computed once into LDS and broadcast across all 128 k-threads.
// Padding mask folded in as dist=+inf -> exp(-inf)=0.
// ---------------------------------------------------------------------------
__global__ __launch_bounds__(K_) void gaussian_sum_kernel(
    const float* __restrict__ pos, const unsigned char* __restrict__ padding_mask,
    const float* __restrict__ means, const float* __restrict__ stds,
    float* __restrict__ sum_pf)
{
    __shared__ float dist_s[N_];
    const int row = blockIdx.x;        // b*N + i
    const int b   = row / N_;
    const int tid = threadIdx.x;

    const float xi = pos[row * 3 + 0];
    const float yi = pos[row * 3 + 1];
    const float zi = pos[row * 3 + 2];

    for (int j = tid; j < N_; j += K_) {
        const int jr = b * N_ + j;
        const float dx = xi - pos[jr * 3 + 0];
        const float dy = yi - pos[jr * 3 + 1];
        const float dz = zi - pos[jr * 3 + 2];
        const float d  = sqrtf(dx * dx + dy * dy + dz * dz);
        dist_s[j] = padding_mask[jr] ? __int_as_float(0x7f800000) : d; // +inf kills term
    }
    __syncthreads();

    const float mean   = means[tid];
    const float sd     = fabsf(stds[tid]) + 0.01f;
    const float inv_sd = 1.0f / sd;
    const float scale  = inv_sd * rsqrtf(2.0f * 3.14159f); // reference PI constant

    float acc = 0.0f;
#pragma unroll 4
    for (int j = 0; j < N_; ++j) {
        const float z = (dist_s[j] - mean) * inv_sd;
        acc += __expf(-0.5f * z * z);   // v_exp_f32
    }
    sum_pf[row * K_ + tid] = acc * scale;
}

// ---------------------------------------------------------------------------
// Kernel 2: fused node MLP using fp32 WMMA (V_WMMA_F32_16X16X4_F32).
//   H   = gelu(sum_pf[16,128] @ W1[128,128])
//   out[...,0:256] = H @ W2[128,256]
// 96 blocks (16-row tiles), 256 threads = 8 waves. Wave w owns 16 columns of
// GEMM1 and 2x16 columns of GEMM2. LDS rows padded +4 to avoid bank conflicts.
// ---------------------------------------------------------------------------
__global__ __launch_bounds__(256) void node_mlp_kernel(
    const float* __restrict__ sum_pf, const float* __restrict__ w1,
    const float* __restrict__ w2, float* __restrict__ out)
{
    __shared__ float A[16][K_ + 4];
    __shared__ float H[16][K_ + 4];

    const int tile = blockIdx.x;       // 0..95
    const int tid  = threadIdx.x;
    const int lane = tid & 31;
    const int wid  = tid >> 5;         // 0..7 (wave32)

    // Stage 16x128 A tile
    for (int idx = tid; idx < 16 * K_; idx += 256) {
        const int m = idx >> 7, k = idx & (K_ - 1);
        A[m][k] = sum_pf[(tile * 16 + m) * K_ + k];
    }
    __syncthreads();

    const int m    = lane & 15;             // A-row / B-col within fragment
    const int khi  = (lane >> 4) << 1;      // 0 for lanes 0-15, 2 for 16-31
    const int mhi  = (lane >> 4) << 3;      // 0 or 8 for C/D rows

    // ---- GEMM1: H = gelu(A @ W1), wave wid -> columns [wid*16, wid*16+16)
    {
        v8f acc = {};
        const float* w1col = w1 + wid * 16 + m;   // W1[k][n], n = wid*16 + m
        for (int kk = 0; kk < K_; kk += 4) {
            const int kb = kk + khi;
            v2f a;  a.x = A[m][kb];          a.y = A[m][kb + 1];
            v2f bb; bb.x = w1col[kb * K_];   bb.y = w1col[(kb + 1) * K_];
            acc = __builtin_amdgcn_wmma_f32_16x16x4_f32(
                false, a, false, bb, (short)0, acc, false, false);
        }
#pragma unroll
        for (int r = 0; r < 8; ++r) {
            H[r + mhi][wid * 16 + m] = gelu_erf(acc[r]);
        }
    }
    __syncthreads();

    // ---- GEMM2: out = H @ W2, wave wid -> column tiles (wid*2+t)*16
#pragma unroll
    for (int t = 0; t < 2; ++t) {
        const int cb = (wid * 2 + t) * 16;
        v8f acc = {};
        const float* w2col = w2 + cb + m;         // W2[k][n], n = cb + m
        for (int kk = 0; kk < K_; kk += 4) {
            const int kb = kk + khi;
            v2f a;  a.x = H[m][kb];             a.y = H[m][kb + 1];
            v2f bb; bb.x = w2col[kb * INTER_];  bb.y = w2col[(kb + 1) * INTER_];
            acc = __builtin_amdgcn_wmma_f32_16x16x4_f32(
                false, a, false, bb, (short)0, acc, false, false);
        }
#pragma unroll
        for (int r = 0; r < 8; ++r) {
            out[(tile * 16 + r + mhi) * E_ + cb + m] = acc[r];
        }
    }
}

// ---------------------------------------------------------------------------
// Kernel 3: time-embedding MLP. 3 blocks: row 0/1 -> te[b], row 2 -> t0e.
// ---------------------------------------------------------------------------
__global__ __launch_bounds__(E_) void time_mlp_kernel(
    const int* __restrict__ time_pos,
    const float* __restrict__ w1, const float* __restrict__ b1,
    const float* __restrict__ w2, const float* __restrict__ b2,
    float* __restrict__ time_rows)
{
    __shared__ float emb[E_];
    __shared__ float hid[E_];
    const int rowi = blockIdx.x;
    const int n    = threadIdx.x;
    const float t  = (rowi < B_) ? (float)time_pos[rowi] : 0.0f;
    const int half = E_ / 2;

    {
        const int i   = (n < half) ? n : (n - half);
        const float f = expf(-logf(10000.0f) * (float)i / (float)half);
        emb[n] = (n < half) ? sinf(t * f) : cosf(t * f);
    }
    __syncthreads();

    float s = b1[n];
    for (int i = 0; i < E_; ++i) s += emb[i] * w1[i * E_ + n];
    hid[n] = s / (1.0f + expf(-s));        // SiLU
    __syncthreads();

    float o = b2[n];
    for (int i = 0; i < E_; ++i) o += hid[i] * w2[i * E_ + n];
    time_rows[rowi * E_ + n] = o;
}

// ---------------------------------------------------------------------------
// Kernel 4: angle branch (writes out[...,256:512]) + time-embedding add.
// One block per (b,n) row, 512 threads (one per output channel).
// ---------------------------------------------------------------------------
__global__ __launch_bounds__(E_) void tail_kernel(
    const float* __restrict__ angle, const unsigned char* __restrict__ mask_pos,
    const float* __restrict__ aw1, const float* __restrict__ aw2,
    const float* __restrict__ time_rows, float* __restrict__ out)
{
    const int row = blockIdx.x;
    const int e   = threadIdx.x;
    const int b   = row / N_;
    const float tv = mask_pos[row] ? time_rows[b * E_ + e] : time_rows[2 * E_ + e];

    if (e < INTER_) {
        out[row * E_ + e] += tv;           // node3d already there
    } else {
        float a0 = angle[row * 3 + 0];
        float a1 = angle[row * 3 + 1];
        float a2 = angle[row * 3 + 2];
        a0 = (a0 == INFINITY) ? 0.0f : a0; // isposinf -> 0
        a1 = (a1 == INFINITY) ? 0.0f : a1;
        a2 = (a2 == INFINITY) ? 0.0f : a2;
        const float g0 = gelu_erf(a0 * aw1[0] + a1 * aw1[3] + a2 * aw1[6]);
        const float g1 = gelu_erf(a0 * aw1[1] + a1 * aw1[4] + a2 * aw1[7]);
        const float g2 = gelu_erf(a0 * aw1[2] + a1 * aw1[5] + a2 * aw1[8]);
        const int j = e - INTER_;
        out[row * E_ + e] =
            g0 * aw2[0 * INTER_ + j] + g1 * aw2[1 * INTER_ + j] + g2 * aw2[2 * INTER_ + j] + tv;
    }
}

// ---------------------------------------------------------------------------
extern "C" void kernel_launch(void* const* d_in, const int* in_sizes, int n_in,
                              void* d_out, int out_size, void* d_ws, size_t ws_size,
                              hipStream_t stream) {
    const float*         pos          = (const float*)d_in[0];
    const float*         angle        = (const float*)d_in[1];
    /* d_in[2] node_type_edge: unused by reference */
    const unsigned char* padding_mask = (const unsigned char*)d_in[3];
    /* d_in[4] mask_aa: unused by reference */
    const unsigned char* mask_pos     = (const unsigned char*)d_in[5];
    const int*           time_pos     = (const int*)d_in[6];
    const float*         means        = (const float*)d_in[7];
    const float*         stds         = (const float*)d_in[8];
    const float*         fp_w1        = (const float*)d_in[9];
    const float*         fp_w2        = (const float*)d_in[10];
    const float*         ang_w1       = (const float*)d_in[11];
    const float*         ang_w2       = (const float*)d_in[12];
    const float*         t_w1         = (const float*)d_in[13];
    const float*         t_b1         = (const float*)d_in[14];
    const float*         t_w2         = (const float*)d_in[15];
    const float*         t_b2         = (const float*)d_in[16];

    float* out       = (float*)d_out;
    float* sum_pf    = (float*)d_ws;                 // ROWS_*K_ floats
    float* time_rows = sum_pf + (size_t)ROWS_ * K_;  // 3*E_ floats

    gaussian_sum_kernel<<<ROWS_, K_, 0, stream>>>(pos, padding_mask, means, stds, sum_pf);
    time_mlp_kernel<<<3, E_, 0, stream>>>(time_pos, t_w1, t_b1, t_w2, t_b2, time_rows);
    node_mlp_kernel<<<ROWS_ / 16, 256, 0, stream>>>(sum_pf, fp_w1, fp_w2, out);
    tail_kernel<<<ROWS_, E_, 0, stream>>>(angle, mask_pos, ang_w1, ang_w2, time_rows, out);
}